// MoeMLP_64398739636441
// MI455X (gfx1250) — compile-verified
//
#include <hip/hip_runtime.h>

// ---------------------------------------------------------------------------
// MoE low-rank MLP for MI455X (gfx1250), fp32 via V_WMMA_F32_16X16X4_F32.
// Rationale: reference is fp32 end-to-end with K=2048 reductions; rank R=16
// maps exactly onto the 16x16 WMMA tile. Problem is ~64 GFLOP vs ~143 MB of
// HBM traffic (~450 FLOP/byte) -> feed the matrix pipes, keep fp32 exactness.
// ---------------------------------------------------------------------------

typedef __attribute__((ext_vector_type(2))) float v2f;   // WMMA f32 A/B frag
typedef __attribute__((ext_vector_type(8))) float v8f;   // WMMA f32 C/D tile

#define H_DIM 2048
#define F_DIM 8192
#define R_DIM 16
#define E_NUM 8
#define NEG_SLOPE 0.01f
#define WPB 4            // waves per block; each wave owns a 16-token tile

// D = A(16x4 f32) * B(4x16 f32) + C(16x16 f32)
// (8-arg signature per compile-probe: neg_a, A, neg_b, B, c_mod, C, reuse_a,
//  reuse_b. If the device pass lacks this builtin we get a clear diagnostic
//  at this call site — do NOT preprocessor-guard it; the host pass misfires.)
static __device__ __forceinline__ v8f wmma4(v2f a, v2f b, v8f c) {
  return __builtin_amdgcn_wmma_f32_16x16x4_f32(
      /*neg_a=*/false, a, /*neg_b=*/false, b,
      /*c_mod=*/(short)0, c, /*reuse_a=*/false, /*reuse_b=*/false);
}

// ---------------------------------------------------------------------------
// Router: logits = x @ gate_w^T, fp32 softmax, top-2, renormalize.
// One wave per token; lanes stride over H, shuffle-reduce 8 partial dots.
// Writes dense per-expert combine weights w_te[T,8] (0 for unselected) and
// the tuple output rw[T,2] (top-1 then top-2, renormalized).
// ---------------------------------------------------------------------------
__global__ __launch_bounds__(128) void moe_router(
    const float* __restrict__ x, const float* __restrict__ gate_w,
    float* __restrict__ w_te, float* __restrict__ rw_out, int T) {
  int gtid = blockIdx.x * blockDim.x + threadIdx.x;
  int tok  = gtid >> 5;
  int lane = gtid & 31;
  if (tok >= T) return;

  const float* xr = x + (size_t)tok * H_DIM;
  float acc[E_NUM];
#pragma unroll
  for (int e = 0; e < E_NUM; ++e) acc[e] = 0.f;

  for (int h = lane; h < H_DIM; h += 32) {
    float xv = xr[h];
#pragma unroll
    for (int e = 0; e < E_NUM; ++e) acc[e] += xv * gate_w[e * H_DIM + h];
  }
#pragma unroll
  for (int off = 16; off > 0; off >>= 1) {
#pragma unroll
    for (int e = 0; e < E_NUM; ++e) acc[e] += __shfl_down(acc[e], off, 32);
  }

  if (lane == 0) {
    float m = acc[0];
#pragma unroll
    for (int e = 1; e < E_NUM; ++e) m = fmaxf(m, acc[e]);
    float p[E_NUM];
#pragma unroll
    for (int e = 0; e < E_NUM; ++e) p[e] = __expf(acc[e] - m);
    // top-2 (ties -> lower index first, matching jax.lax.top_k)
    int i1 = 0;
#pragma unroll
    for (int e = 1; e < E_NUM; ++e) if (p[e] > p[i1]) i1 = e;
    int i2 = (i1 == 0) ? 1 : 0;
#pragma unroll
    for (int e = 0; e < E_NUM; ++e)
      if (e != i1 && e != i2 && p[e] > p[i2]) i2 = e;
    float s  = p[i1] + p[i2];
    float r1 = p[i1] / s;
    float r2 = p[i2] / s;
#pragma unroll
    for (int e = 0; e < E_NUM; ++e)
      w_te[(size_t)tok * E_NUM + e] = (e == i1) ? r1 : ((e == i2) ? r2 : 0.f);
    rw_out[(size_t)tok * 2 + 0] = r1;
    rw_out[(size_t)tok * 2 + 1] = r2;
  }
}

// ---------------------------------------------------------------------------
// Expert kernel. Each wave owns 16 tokens. Per expert:
//   a1 = X w1A^T, a3 = X w3A^T      (WMMA K-loop over H)
//   for each 16-wide f tile: h1 = a1 w1B^T, h3 = a3 w3B^T,
//        h = leaky(h1)*h3, b += h w2A^T    (LDS bounce for C/D -> A layout)
// then y = sum_e w_te[:,e] * (b_e w2B^T) tiled over H, single store.
//
// WMMA f32 operand mapping assumed (per ISA 16x4 A table, mirrored for B):
//   A frag (16x4): lanes 0-15 row M=lane hold {K=k0,k0+1}; lanes 16-31 row
//                  M=lane-16 hold {K=k0+2,k0+3}  -> per-lane contiguous pair.
//   B frag (4x16): lane column N=lane%16, pair {K=k0+2*(lane/16), +1}.
//   C/D (16x16):   VGPR v = row v + 8*(lane/16), column N = lane%16.
// All per-wave LDS regions are private -> same-wave DS ordering, no barriers.
// ---------------------------------------------------------------------------
#define LDSF_A1 0
#define LDSF_A3 256
#define LDSF_H  512
#define LDSF_B  768                 // 8 experts * 256
#define LDSF_PER_WAVE 2816          // floats

__global__ __launch_bounds__(WPB * 32) void moe_experts(
    const float* __restrict__ x,
    const float* __restrict__ w1A, const float* __restrict__ w1B,
    const float* __restrict__ w2A, const float* __restrict__ w2B,
    const float* __restrict__ w3A, const float* __restrict__ w3B,
    const float* __restrict__ w_te, float* __restrict__ out) {
  __shared__ float lds[WPB * LDSF_PER_WAVE];

  const int lane = threadIdx.x & 31;
  const int wave = threadIdx.x >> 5;
  const int lo   = lane & 15;       // column / N index
  const int hi   = lane >> 4;       // lane-half
  const int k0   = 2 * hi;          // K offset of this lane's frag pair
  const int rowA = lo;              // A-operand row for this lane
  const int rowC = 8 * hi;          // C/D row base for this lane
  const int t0   = (blockIdx.x * WPB + wave) * 16;

  float* myLds = lds + wave * LDSF_PER_WAVE;
  float* aBuf1 = myLds + LDSF_A1;
  float* aBuf3 = myLds + LDSF_A3;
  float* hBuf  = myLds + LDSF_H;
  float* bBuf  = myLds + LDSF_B;    // [E][16][16]

  for (int e = 0; e < E_NUM; ++e) {
    // ---- Stage 1: rank projections a1, a3  (M=16, N=16, K=H) --------------
    v8f acc1 = {};
    v8f acc3 = {};
    const float* xA  = x   + (size_t)(t0 + rowA) * H_DIM + k0;
    const float* b1p = w1A + (size_t)e * R_DIM * H_DIM + (size_t)lo * H_DIM + k0;
    const float* b3p = w3A + (size_t)e * R_DIM * H_DIM + (size_t)lo * H_DIM + k0;
#pragma unroll 4
    for (int k = 0; k < H_DIM; k += 4) {
      v2f a  = *(const v2f*)(xA + k);
      v2f b1 = *(const v2f*)(b1p + k);
      v2f b3 = *(const v2f*)(b3p + k);
      acc1 = wmma4(a, b1, acc1);
      acc3 = wmma4(a, b3, acc3);
    }
    // C/D layout -> row-major LDS (layout bounce for reuse as A operand)
#pragma unroll
    for (int v = 0; v < 8; ++v) {
      aBuf1[(v + rowC) * 16 + lo] = acc1[v];
      aBuf3[(v + rowC) * 16 + lo] = acc3[v];
    }
    v2f a1f[4], a3f[4];
#pragma unroll
    for (int j = 0; j < 4; ++j) {
      a1f[j] = *(const v2f*)&aBuf1[rowA * 16 + 4 * j + k0];
      a3f[j] = *(const v2f*)&aBuf3[rowA * 16 + 4 * j + k0];
    }

    // ---- Stage 2: F loop, fused expand -> activation -> contract ----------
    v8f accB = {};
    const float* w1Bp = w1B + (size_t)e * F_DIM * R_DIM + (size_t)lo * R_DIM + k0;
    const float* w3Bp = w3B + (size_t)e * F_DIM * R_DIM + (size_t)lo * R_DIM + k0;
    const float* w2Ap = w2A + (size_t)e * R_DIM * F_DIM + (size_t)lo * F_DIM + k0;
    for (int f = 0; f < F_DIM; f += 16) {
      const float* w1Bf = w1Bp + (size_t)f * R_DIM;
      const float* w3Bf = w3Bp + (size_t)f * R_DIM;
      v8f h1 = {};
      v8f h3 = {};
#pragma unroll
      for (int j = 0; j < 4; ++j) {
        h1 = wmma4(a1f[j], *(const v2f*)(w1Bf + 4 * j), h1);
        h3 = wmma4(a3f[j], *(const v2f*)(w3Bf + 4 * j), h3);
      }
      // h = leaky_relu(h1) * h3, staged through LDS for A-operand reload
#pragma unroll
      for (int v = 0; v < 8; ++v) {
        float z = h1[v];
        float l = (z >= 0.f) ? z : NEG_SLOPE * z;
        hBuf[(v + rowC) * 16 + lo] = l * h3[v];
      }
      const float* w2Af = w2Ap + f;
#pragma unroll
      for (int j = 0; j < 4; ++j) {
        v2f hA = *(const v2f*)&hBuf[rowA * 16 + 4 * j + k0];
        accB = wmma4(hA, *(const v2f*)(w2Af + 4 * j), accB);
      }
    }
    // stash b_e tile (row-major) for stage 3
#pragma unroll
    for (int v = 0; v < 8; ++v)
      bBuf[e * 256 + (v + rowC) * 16 + lo] = accB[v];
  }

  // preload per-row router weights for this lane's C/D rows
  float wreg[E_NUM][8];
#pragma unroll
  for (int e = 0; e < E_NUM; ++e)
#pragma unroll
    for (int v = 0; v < 8; ++v)
      wreg[e][v] = w_te[(size_t)(t0 + v + rowC) * E_NUM + e];

  // ---- Stage 3: out = sum_e w_te[:,e] * (b_e @ w2B[e]^T), tiled over H ----
  for (int n0 = 0; n0 < H_DIM; n0 += 16) {
    v8f accY = {};
#pragma unroll
    for (int e = 0; e < E_NUM; ++e) {
      v2f bA[4];
#pragma unroll
      for (int j = 0; j < 4; ++j)
        bA[j] = *(const v2f*)&bBuf[e * 256 + rowA * 16 + 4 * j + k0];
      const float* w2Bp =
          w2B + (size_t)e * H_DIM * R_DIM + (size_t)(n0 + lo) * R_DIM + k0;
      v8f yb = {};
#pragma unroll
      for (int j = 0; j < 4; ++j)
        yb = wmma4(bA[j], *(const v2f*)(w2Bp + 4 * j), yb);
#pragma unroll
      for (int v = 0; v < 8; ++v) accY[v] += wreg[e][v] * yb[v];
    }
#pragma unroll
    for (int v = 0; v < 8; ++v)
      out[(size_t)(t0 + v + rowC) * H_DIM + n0 + lo] = accY[v];
  }
}

// ---------------------------------------------------------------------------
// Host launcher. Inputs: x, gate_w, w1_A, w1_B, w2_A, w2_B, w3_A, w3_B (f32).
// d_out = [out (T*H floats), rw (T*2 floats)] concatenated (tuple return).
// Workspace: w_te dense combine weights, T*E*4 = 256 KB.
// ---------------------------------------------------------------------------
extern "C" void kernel_launch(void* const* d_in, const int* in_sizes, int n_in,
                              void* d_out, int out_size, void* d_ws,
                              size_t ws_size, hipStream_t stream) {
  const float* x      = (const float*)d_in[0];
  const float* gate_w = (const float*)d_in[1];
  const float* w1A    = (const float*)d_in[2];
  const float* w1B    = (const float*)d_in[3];
  const float* w2A    = (const float*)d_in[4];
  const float* w2B    = (const float*)d_in[5];
  const float* w3A    = (const float*)d_in[6];
  const float* w3B    = (const float*)d_in[7];

  const int T = in_sizes[0] / H_DIM;   // B*S tokens (8192)

  float* out  = (float*)d_out;
  float* rw   = out + (size_t)T * H_DIM;
  float* w_te = (float*)d_ws;          // T*E floats

  // Router: 1 wave per token, 4 waves per block.
  moe_router<<<(T + 3) / 4, 128, 0, stream>>>(x, gate_w, w_te, rw, T);

  // Experts: 4 waves/block, 16 tokens/wave -> 64 tokens/block.
  moe_experts<<<T / (WPB * 16), WPB * 32, 0, stream>>>(
      x, w1A, w1B, w2A, w2B, w3A, w3B, w_te, out);
}